// TIME_WARPING_16698832847537
// MI455X (gfx1250) — compile-verified
//
#include <hip/hip_runtime.h>
#include <hip/hip_bf16.h>

typedef __attribute__((ext_vector_type(2))) float v2f;
typedef __attribute__((ext_vector_type(8))) float v8f;
typedef __attribute__((ext_vector_type(4))) unsigned int v4u;
typedef __attribute__((ext_vector_type(8))) int v8i;
typedef __attribute__((ext_vector_type(4))) int v4i;

#define SLEN 4096   // S
#define NSIG 8192   // B*C
#define NCH  64     // C
#define BAND 16     // half-bandwidth of A^-1 at fp32 precision (|mu|^17 ~ 2e-10)
#define NSLICE 12   // 48 / 4  (K-slices per 16-wide output tile)
#define NT   4      // position tiles per block
#define POS  64     // positions per block = NT*16
#define RW   96     // rhs window  = POS + 2*BAND
#define XW   98     // x   window  = RW + 2 (stencil halo)

// ---------------------------------------------------------------------------
// Kernel 1: M = A^-1 * rhs via banded-Toeplitz convolution cast as WMMA GEMM.
// One wave per block; TDM stages the x tile (16 signals x 98 floats, stride S)
// into LDS, rhs is formed in LDS, then 4 output tiles of
// D(16x16) = W(16x48, Toeplitz g) x rhs(48x16) via chained 16x16x4 f32 WMMAs.
// ---------------------------------------------------------------------------
__global__ __launch_bounds__(32) void tw_spline_M_wmma(
    const float* __restrict__ x, float* __restrict__ Mout) {
  __shared__ float x_lds[16 * XW];     // [sig][u], u = j - jlo
  __shared__ float rhs_lds[RW * 16];   // [q][sig], q = j - (i0-16)
  __shared__ float gtab[64];           // g[d+31], zero for |d| > BAND
  __shared__ float dtile[16 * 17];     // store bounce, padded

  const int lane = threadIdx.x;        // 0..31 (wave32)
  const int i0 = blockIdx.x * POS;     // strip start position
  const int s0 = blockIdx.y * 16;      // signal tile start
  const int jlo = (i0 >= 17) ? (i0 - 17) : 0;
  const int sh = i0 - 17 - jlo;        // 0 interior, -17 for the head strip

  // ---- Tensor Data Mover: async 2D tile x[s0..s0+16)[jlo..jlo+XW) -> LDS ----
  {
    unsigned lds_base = (unsigned)(uintptr_t)(&x_lds[0]);
    unsigned long long ga =
        (unsigned long long)(uintptr_t)(x + (size_t)s0 * SLEN + jlo);
    unsigned td0 = (unsigned)(SLEN - jlo);  // OOB reads past signal end -> 0
    v4u g0;
    g0.x = 1u;                                        // count=1 (valid user D#)
    g0.y = lds_base;                                  // lds_addr
    g0.z = (unsigned)(ga & 0xFFFFFFFFu);              // global_addr[31:0]
    g0.w = (unsigned)((ga >> 32) & 0x01FFFFFFu) | (2u << 30);  // addr[56:32], type=2
    v8i g1;
    g1[0] = (int)(2u << 16);                          // data_size = 4B; mask=0
    g1[1] = (int)((td0 & 0xFFFFu) << 16);             // tensor_dim0[15:0]
    g1[2] = (int)(((td0 >> 16) & 0xFFFFu) | (16u << 16)); // dim0 hi | tensor_dim1=16
    g1[3] = (int)((unsigned)XW << 16);                // tile_dim0 = 98
    g1[4] = 16;                                       // tile_dim1 = 16, tile_dim2 = 0
    g1[5] = SLEN;                                     // tensor_dim0_stride lo
    g1[6] = 0;
    g1[7] = 0;
    v4i gz4 = (v4i){0, 0, 0, 0};                      // groups 2/3 unused (2D)
    v8i gz8 = (v8i){0, 0, 0, 0, 0, 0, 0, 0};          // extra group unused
    __builtin_amdgcn_tensor_load_to_lds(g0, g1, gz4, gz4, gz8, 0);
  }

  // Build Toeplitz kernel table while the TDM runs.
  if (lane == 0) {
    const float mu = -0.26794919243112270f;  // sqrt(3) - 2
    float g = 0.28867513459481287f;          // 1 / (2*sqrt(3))
    for (int q = 0; q < 64; ++q) gtab[q] = 0.0f;
    gtab[31] = g;
    for (int k = 1; k <= BAND; ++k) { g *= mu; gtab[31 + k] = g; gtab[31 - k] = g; }
  }
  __builtin_amdgcn_s_wait_tensorcnt(0);
  __syncthreads();

  // rhs_q = 6*(x_{j-1} - 2x_j + x_{j+1}) for j = i0-16+q, zero at j=0, S-1
  // and outside the tensor (zero-pad). x_lds tap index u = q + sh (+0,+1,+2).
  for (int it = 0; it < (RW * 16) / 32; ++it) {
    int p = it * 32 + lane;
    int q = p % RW;
    int sig = p / RW;
    int j = i0 - BAND + q;
    float r = 0.0f;
    if (j >= 1 && j <= SLEN - 2) {
      int u = sig * XW + q + sh;
      r = 6.0f * (x_lds[u] - 2.0f * x_lds[u + 1] + x_lds[u + 2]);
    }
    rhs_lds[q * 16 + sig] = r;
  }
  __syncthreads();

  // WMMA operand layout (f32 16x16x4, wave32):
  //  A (16x4): lanes 0-15 hold row M=L for K=0(v0),1(v1); lanes 16-31: K=2,3.
  //  B (4x16): lanes 0-15 hold col N=L for K=0(v0),1(v1); lanes 16-31: K=2,3.
  const int posL = lane & 15;
  const int khalf = (lane >> 4) << 1;
  // A weights are position-tile invariant: W[pos, j] = g[j - pos].
  v2f aw[NSLICE];
#pragma unroll
  for (int sl = 0; sl < NSLICE; ++sl) {
    int jb = sl * 4 + khalf;
    aw[sl].x = gtab[jb - BAND - posL + 31];
    aw[sl].y = gtab[jb + 1 - BAND - posL + 31];
  }

  const int rowoff = (lane >> 4) * 8;
  for (int t = 0; t < NT; ++t) {
    v8f acc = {};
#pragma unroll
    for (int sl = 0; sl < NSLICE; ++sl) {
      int jb = t * 16 + sl * 4 + khalf;
      v2f b;
      b.x = rhs_lds[jb * 16 + posL];
      b.y = rhs_lds[(jb + 1) * 16 + posL];
      acc = __builtin_amdgcn_wmma_f32_16x16x4_f32(false, aw[sl], false, b,
                                                  (short)0, acc, false, false);
    }
    // D layout: VGPR r, lanes 0-15: D[M=r, N=L]; lanes 16-31: D[M=r+8, N=L-16].
#pragma unroll
    for (int r = 0; r < 8; ++r) dtile[(rowoff + r) * 17 + posL] = acc[r];
    __syncthreads();
    int i0t = i0 + t * 16;
#pragma unroll
    for (int it = 0; it < 2; ++it) {
      int v = it * 32 + lane;
      int sig = v >> 2;
      int p4 = (v & 3) * 4;
      float4 val;
      val.x = dtile[(p4 + 0) * 17 + sig];
      val.y = dtile[(p4 + 1) * 17 + sig];
      val.z = dtile[(p4 + 2) * 17 + sig];
      val.w = dtile[(p4 + 3) * 17 + sig];
      *reinterpret_cast<float4*>(Mout + (size_t)(s0 + sig) * SLEN + i0t + p4) = val;
    }
    __syncthreads();
  }
}

// ---------------------------------------------------------------------------
// Kernel 2: exact not-a-knot boundary fix. The convolution is fp32-exact for
// i in [~20, S-21]; overwrite M[0..63] and M[S-64..S-1] with exact small
// Thomas solves anchored on the (exact) interior values M[64] / M[S-65].
// One lane per signal.
// ---------------------------------------------------------------------------
__global__ void tw_spline_M_boundary(const float* __restrict__ x,
                                     float* __restrict__ M) {
  int s = blockIdx.x * blockDim.x + threadIdx.x;
  if (s >= NSIG) return;
  const float* xs = x + (size_t)s * SLEN;
  float* Ms = M + (size_t)s * SLEN;

  float r[64];
  float d[64];

  // ---- HEAD: unknowns M0..M63, known M64; row0: M0 - 2M1 + M2 = 0 ----
  r[0] = 0.0f;
  for (int i = 1; i < 64; ++i)
    r[i] = 6.0f * (xs[i - 1] - 2.0f * xs[i] + xs[i + 1]);
  r[63] -= Ms[64];
  // row1 - row0 => 6*M1 = r1  (both M0 and M2 cancel)
  float M1 = r[1] * (1.0f / 6.0f);
  d[2] = 4.0f;
  r[2] -= M1;
  for (int i = 3; i < 64; ++i) {
    float w = 1.0f / d[i - 1];
    d[i] = 4.0f - w;
    r[i] -= w * r[i - 1];
  }
  float mn = r[63] / d[63];
  Ms[63] = mn;
  for (int i = 62; i >= 2; --i) {
    float mi = (r[i] - mn) / d[i];
    Ms[i] = mi;
    mn = mi;
  }
  Ms[1] = M1;
  Ms[0] = 2.0f * M1 - mn;   // row0: M0 = 2*M1 - M2

  // ---- TAIL: unknowns M[S-64..S-1], known M[S-65];
  //      last row: M[S-3] - 2M[S-2] + M[S-1] = 0 ----
  float mknown = Ms[SLEN - 65];
  for (int i = 0; i < 63; ++i) {
    int g = SLEN - 64 + i;
    r[i] = 6.0f * (xs[g - 1] - 2.0f * xs[g] + xs[g + 1]);
  }
  r[0] -= mknown;
  d[0] = 4.0f;
  for (int i = 1; i < 63; ++i) {
    float w = 1.0f / d[i - 1];
    d[i] = 4.0f - w;
    r[i] -= w * r[i - 1];
  }
  // Eliminate last row [.., 1, -2, 1] against bidiagonal rows 61, 62:
  float q = (1.0f / d[61] + 2.0f) / d[62];
  float M63 = (q * r[62] - r[61] / d[61]) / (q + 1.0f);
  float M62 = (r[62] - M63) / d[62];
  Ms[SLEN - 1] = M63;
  Ms[SLEN - 2] = M62;
  mn = M62;
  for (int i = 61; i >= 0; --i) {
    float mi = (r[i] - mn) / d[i];
    Ms[SLEN - 64 + i] = mi;
    mn = mi;
  }
}

// ---------------------------------------------------------------------------
// Kernel 3: warped cubic evaluation + mask select. Gathers are near-contiguous
// (idx ~ i*scale, scale in [0.7,1.3]); mask branch is block-uniform.
// ---------------------------------------------------------------------------
__global__ void tw_warp_eval(const float* __restrict__ x,
                             const float* __restrict__ M,
                             const float* __restrict__ scale,
                             const int* __restrict__ mask,
                             float* __restrict__ out) {
  size_t gid = (size_t)blockIdx.x * blockDim.x + threadIdx.x;
  int i = (int)(gid & (SLEN - 1));
  size_t row = gid >> 12;            // b*C + c
  int b = (int)(row >> 6);           // / NCH
  const float* xs = x + (row << 12);
  float res = xs[i];
  if (mask[b]) {
    float w = fminf(fmaxf((float)i * scale[b], 0.0f), (float)(SLEN - 1));
    int idx = (int)floorf(w);
    idx = max(0, min(idx, SLEN - 2));
    float t = w - (float)idx;
    const float* Mss = M + (row << 12);
    float y0 = xs[idx], y1 = xs[idx + 1];
    float M0 = Mss[idx], M1v = Mss[idx + 1];
    float bb = (y1 - y0) - (2.0f * M0 + M1v) * (1.0f / 6.0f);
    float cc = 0.5f * M0;
    float dd = (M1v - M0) * (1.0f / 6.0f);
    res = y0 + t * (bb + t * (cc + t * dd));
  }
  out[gid] = res;
}

extern "C" void kernel_launch(void* const* d_in, const int* in_sizes, int n_in,
                              void* d_out, int out_size, void* d_ws, size_t ws_size,
                              hipStream_t stream) {
  (void)in_sizes; (void)n_in; (void)out_size; (void)ws_size;
  const float* x = (const float*)d_in[0];
  const float* scale = (const float*)d_in[1];
  const int* mask = (const int*)d_in[2];
  float* out = (float*)d_out;
  float* M = (float*)d_ws;   // B*C*S floats = 128 MB scratch

  dim3 gA(SLEN / POS, NSIG / 16, 1);
  tw_spline_M_wmma<<<gA, 32, 0, stream>>>(x, M);
  tw_spline_M_boundary<<<NSIG / 256, 256, 0, stream>>>(x, M);
  size_t total = (size_t)NSIG * SLEN;
  tw_warp_eval<<<(unsigned)(total / 256), 256, 0, stream>>>(x, M, scale, mask, out);
}